// CRF_NET_10746008174726
// MI455X (gfx1250) — compile-verified
//
#include <hip/hip_runtime.h>
#include <math.h>

// Problem constants from the reference
#define B_TOT   16384
#define M_SEQ   14
#define HW      128      // H*W = 16*8 (K dim of fused GEMM)
#define L_LAB   27
#define LPAD    32       // labels padded for two 16-wide WMMA N tiles
#define C_CH    5
#define KK      2
#define EMBED_  525
#define BG      32       // batch elements per workgroup
#define ROWS    (BG * M_SEQ)   // 448 (b,m) rows per workgroup
#define NT      (ROWS / 16)    // 28 row tiles
#define RSTRIDE 32             // LDS row stride for emis (no divergent stores)

// workspace layout (floats)
#define WS_WEFF 0        // [32][128]
#define WS_BEFF 4096     // [32]
#define WS_ET   4128     // [27*27] exp(T - Tmax_col), padded region
#define WS_TMAX 4864     // [27]

typedef __attribute__((ext_vector_type(2))) float v2f;
typedef __attribute__((ext_vector_type(8))) float v8f;

// ---------------------------------------------------------------------------
// Stage 1: fold conv into emission weights; precompute exp-shifted transition
// matrix; zero the scalar output accumulator.
//   Weff[l][h*8+w] = sum_{c,ki,kj : 0<=h-ki<=14, 0<=w-kj<=6}
//                      conv_w[c,ki,kj] * W[l, c*105 + (h-ki)*7 + (w-kj)]
//   beff[l]        = sum_c conv_b[c] * sum_{i,j} W[l, c*105 + i*7 + j]
//   tmax[j]        = max_i T[i,j];   eT[i,j] = exp(T[i,j] - tmax[j])
// ---------------------------------------------------------------------------
__global__ __launch_bounds__(256)
void crf_prep_kernel(const float* __restrict__ conv_w,
                     const float* __restrict__ conv_b,
                     const float* __restrict__ W,
                     const float* __restrict__ Tm,
                     float* __restrict__ ws,
                     float* __restrict__ out)
{
    __shared__ float Ts[L_LAB * L_LAB];
    __shared__ float tmx[L_LAB];

    const int tid = threadIdx.x;
    if (tid == 0) out[0] = 0.0f;

    float* weff = ws + WS_WEFF;
    float* beff = ws + WS_BEFF;
    float* eT   = ws + WS_ET;
    float* tmax = ws + WS_TMAX;

    for (int i = tid; i < L_LAB * L_LAB; i += blockDim.x) Ts[i] = Tm[i];
    __syncthreads();

    if (tid < L_LAB) {
        float mx = -3.402823466e38f;
        for (int i = 0; i < L_LAB; ++i) mx = fmaxf(mx, Ts[i * L_LAB + tid]);
        tmx[tid]  = mx;
        tmax[tid] = mx;
    }
    __syncthreads();

    for (int i = tid; i < L_LAB * L_LAB; i += blockDim.x)
        eT[i] = __expf(Ts[i] - tmx[i % L_LAB]);

    for (int idx = tid; idx < LPAD * HW; idx += blockDim.x) {
        const int l   = idx / HW;
        const int pix = idx - l * HW;
        const int h   = pix >> 3;
        const int w   = pix & 7;
        float acc = 0.0f;
        if (l < L_LAB) {
            for (int c = 0; c < C_CH; ++c) {
                for (int ki = 0; ki < KK; ++ki) {
                    const int i = h - ki;
                    if (i < 0 || i > 14) continue;
                    for (int kj = 0; kj < KK; ++kj) {
                        const int j = w - kj;
                        if (j < 0 || j > 6) continue;
                        acc += conv_w[(c * KK + ki) * KK + kj] *
                               W[l * EMBED_ + c * 105 + i * 7 + j];
                    }
                }
            }
        }
        weff[idx] = acc;
    }

    if (tid < LPAD) {
        float acc = 0.0f;
        if (tid < L_LAB) {
            for (int c = 0; c < C_CH; ++c) {
                const float cb = conv_b[c];
                float s = 0.0f;
                for (int e = 0; e < 105; ++e)
                    s += W[tid * EMBED_ + c * 105 + e];
                acc += cb * s;
            }
        }
        beff[tid] = acc;
    }
}

// ---------------------------------------------------------------------------
// Stage 2: fused emission GEMM (fp32 WMMA 16x16x4) + CRF forward recursion.
// One workgroup = 32 batch elements, 256 threads = 8 wave32s.
// ---------------------------------------------------------------------------
__global__ __launch_bounds__(256)
void crf_main_kernel(const float* __restrict__ x,       // [B*M][128]
                     const int*   __restrict__ labels,  // [B][M]
                     const float* __restrict__ Tm,      // [27][27] raw (gold path)
                     const float* __restrict__ ws,
                     float* __restrict__ out)
{
    __shared__ float emis[ROWS * RSTRIDE];   // 57344 B (alpha written in place)
    __shared__ float eTl[L_LAB * L_LAB];     //  2916 B  exp(T - tmax)
    __shared__ float tml[L_LAB];             //   108 B
    __shared__ float ea[BG * 28];            //  3584 B  exp(alpha - mxa)
    __shared__ float mxa[BG];                //   128 B

    const int tid   = threadIdx.x;
    const int wg_b0 = blockIdx.x * BG;

    const float* weff = ws + WS_WEFF;
    const float* beff = ws + WS_BEFF;

    for (int i = tid; i < L_LAB * L_LAB; i += 256) eTl[i] = ws[WS_ET + i];
    if (tid < L_LAB) tml[tid] = ws[WS_TMAX + tid];

    const int wave  = tid >> 5;
    const int lane  = tid & 31;
    const int khalf = lane >> 4;   // K-pair select (A/B fragment layout)
    const int lr    = lane & 15;   // M row (A) / N col (B,C,D)

    const float bias0 = beff[lr];
    const float bias1 = beff[16 + lr];

    const float* w0p = weff + lr * HW + khalf * 2;          // labels 0..15
    const float* w1p = weff + (16 + lr) * HW + khalf * 2;   // labels 16..31

    // --- Phase A: emis[row, 0..31] = X(row,128) * WeffT(128,32) + beff ---
    for (int tile = wave; tile < NT; tile += 8) {
        const float* xr =
            x + ((size_t)wg_b0 * M_SEQ + (size_t)tile * 16 + lr) * HW + khalf * 2;

        v8f c0 = {bias0, bias0, bias0, bias0, bias0, bias0, bias0, bias0};
        v8f c1 = {bias1, bias1, bias1, bias1, bias1, bias1, bias1, bias1};

        #pragma unroll 4
        for (int k0 = 0; k0 < HW; k0 += 4) {
            v2f a  = *(const v2f*)(xr  + k0);
            v2f b0 = *(const v2f*)(w0p + k0);
            v2f b1 = *(const v2f*)(w1p + k0);
            c0 = __builtin_amdgcn_wmma_f32_16x16x4_f32(
                     false, a, false, b0, (short)0, c0, false, false);
            c1 = __builtin_amdgcn_wmma_f32_16x16x4_f32(
                     false, a, false, b1, (short)0, c1, false, false);
        }

        const int row0 = tile * 16 + khalf * 8;  // D layout: M = r + 8*khalf
        #pragma unroll
        for (int r = 0; r < 8; ++r) {
            emis[(row0 + r) * RSTRIDE + lr]      = c0[r];   // labels 0..15
            emis[(row0 + r) * RSTRIDE + 16 + lr] = c1[r];   // labels 16..31
        }
    }
    __syncthreads();

    // --- gold path score (one thread per batch element; raw T from global) ---
    float score = 0.0f;
    if (tid < BG) {
        const int* lb = labels + (size_t)(wg_b0 + tid) * M_SEQ;
        int prev = lb[0];
        score = emis[(tid * M_SEQ) * RSTRIDE + prev];
        for (int m = 1; m < M_SEQ; ++m) {
            const int cur = lb[m];
            score += emis[(tid * M_SEQ + m) * RSTRIDE + cur] +
                     Tm[prev * L_LAB + cur];
            prev = cur;
        }
    }
    __syncthreads();

    // --- forward recursion, alpha in place over emis rows ---
    // lse_i(a_i + T_ij) = mxa + tmax_j + log( sum_i exp(a_i-mxa)*eT[i][j] )
    for (int m = 1; m < M_SEQ; ++m) {
        // (a) per-b row max of alpha
        if (tid < BG) {
            const float* ar = &emis[(tid * M_SEQ + m - 1) * RSTRIDE];
            float mx = -3.402823466e38f;
            #pragma unroll
            for (int i = 0; i < L_LAB; ++i) mx = fmaxf(mx, ar[i]);
            mxa[tid] = mx;
        }
        __syncthreads();
        // (b) ea = exp(alpha - mxa): only 27 exps per (b,m)
        for (int it = tid; it < BG * L_LAB; it += 256) {
            const int bl = it / L_LAB;
            const int i  = it - bl * L_LAB;
            ea[bl * 28 + i] =
                __expf(emis[(bl * M_SEQ + m - 1) * RSTRIDE + i] - mxa[bl]);
        }
        __syncthreads();
        // (c) 27-wide FMA matvec + single log per (b,j)
        for (int it = tid; it < BG * L_LAB; it += 256) {
            const int bl = it / L_LAB;
            const int j  = it - bl * L_LAB;
            float s = 0.0f;
            #pragma unroll
            for (int i = 0; i < L_LAB; ++i)
                s = __builtin_fmaf(ea[bl * 28 + i], eTl[i * L_LAB + j], s);
            emis[(bl * M_SEQ + m) * RSTRIDE + j] += mxa[bl] + tml[j] + __logf(s);
        }
        __syncthreads();
    }

    // --- logZ and output accumulation ---
    if (tid < BG) {
        const float* ar = &emis[(tid * M_SEQ + (M_SEQ - 1)) * RSTRIDE];
        float mx = -3.402823466e38f;
        for (int j = 0; j < L_LAB; ++j) mx = fmaxf(mx, ar[j]);
        float s = 0.0f;
        for (int j = 0; j < L_LAB; ++j) s += __expf(ar[j] - mx);
        const float logZ = mx + __logf(s);
        atomicAdd(out, logZ - score);
    }
}

// ---------------------------------------------------------------------------
extern "C" void kernel_launch(void* const* d_in, const int* in_sizes, int n_in,
                              void* d_out, int out_size, void* d_ws, size_t ws_size,
                              hipStream_t stream)
{
    const float* x      = (const float*)d_in[0];
    const int*   labels = (const int*)  d_in[1];   // jnp.int64 downcasts to i32 w/o x64
    const float* conv_w = (const float*)d_in[2];
    const float* conv_b = (const float*)d_in[3];
    const float* W      = (const float*)d_in[4];
    const float* Tm     = (const float*)d_in[5];
    float* out = (float*)d_out;
    float* ws  = (float*)d_ws;   // ~19.6 KB used

    hipLaunchKernelGGL(crf_prep_kernel, dim3(1), dim3(256), 0, stream,
                       conv_w, conv_b, W, Tm, ws, out);
    hipLaunchKernelGGL(crf_main_kernel, dim3(B_TOT / BG), dim3(256), 0, stream,
                       x, labels, Tm, ws, out);
}